// update_u2_25950192403288
// MI455X (gfx1250) — compile-verified
//
#include <hip/hip_runtime.h>
#include <hip/hip_bf16.h>
#include <math.h>

// ---------------------------------------------------------------------------
// Reference:  u = segment_sum(v, batch)          [16384, 128]
//             h = softplus(u @ W1 + b1) - ln(2)  [16384, 256]
//             out = h @ W2 + b2                  [16384, 1]
// Memory-bound on the 256 MB stream of v (~11 us at 23.3 TB/s).
// MLP done with V_WMMA_F32_16X16X4_F32 (exact f32 -> matches f32 reference).
// ---------------------------------------------------------------------------

#define HIDDEN       128
#define HIDDEN2      256
#define NODES_PER_BLOCK 256
#define SSP_SHIFT    0.6931471805599453f   // ln(2)

typedef __attribute__((ext_vector_type(2))) float v2f;
typedef __attribute__((ext_vector_type(8))) float v8f;

// ---------------------------------------------------------------------------
// Kernel 0: zero the u accumulator in workspace.
// ---------------------------------------------------------------------------
__global__ void zero_f32_kernel(float* __restrict__ p, int n) {
    int i = blockIdx.x * blockDim.x + threadIdx.x;
    if (i < n) p[i] = 0.0f;
}

// ---------------------------------------------------------------------------
// Kernel 1: segment-sum readout.  batch is sorted, so runs are contiguous.
// Block = 128 threads = 4 waves. Each block owns a 256-node contiguous slab.
// Wave w walks nodes base+w, base+w+4, ...  Lane holds a float4 column slice
// (cols 4*lane .. 4*lane+3) accumulated in registers; flush via atomicAdd
// only when the segment id changes (boundaries + wave interleave).
// ---------------------------------------------------------------------------
__global__ __launch_bounds__(128)
void segsum_kernel(const float* __restrict__ v,
                   const int*   __restrict__ batch,
                   float*       __restrict__ u,
                   int n_nodes) {
    const int lane = threadIdx.x & 31;
    const int wave = threadIdx.x >> 5;          // 0..3
    const int col  = lane * 4;                  // 32 lanes * 4 = 128 cols

    const int base = blockIdx.x * NODES_PER_BLOCK;
    int end = base + NODES_PER_BLOCK;
    if (end > n_nodes) end = n_nodes;

    float ax = 0.f, ay = 0.f, az = 0.f, aw = 0.f;
    int cur = -1;

    for (int i = base + wave; i < end; i += 4) {
        const int seg = batch[i];
        if (seg != cur) {
            if (cur >= 0) {
                float* up = u + (size_t)cur * HIDDEN + col;
                atomicAdd(up + 0, ax);
                atomicAdd(up + 1, ay);
                atomicAdd(up + 2, az);
                atomicAdd(up + 3, aw);
            }
            cur = seg;
            ax = ay = az = aw = 0.f;
        }
        const float4 val = *(const float4*)(v + (size_t)i * HIDDEN + col);
        ax += val.x; ay += val.y; az += val.z; aw += val.w;
    }
    if (cur >= 0) {
        float* up = u + (size_t)cur * HIDDEN + col;
        atomicAdd(up + 0, ax);
        atomicAdd(up + 1, ay);
        atomicAdd(up + 2, az);
        atomicAdd(up + 3, aw);
    }
}

// ---------------------------------------------------------------------------
// Kernel 2: fused MLP using V_WMMA_F32_16X16X4_F32.
// Block = 512 threads = 16 waves; block handles 16 rows (one M-tile) of u.
// Wave w computes the 16x16 tile of (u @ W1) for columns n0 = 16*w.
//   A (16x4 f32, 2 VGPR): lane<16 -> row M=lane, (K, K+1); lane>=16 -> (K+2,K+3)
//   B (4x16 f32, 2 VGPR): mirrored per-K-row striped across lanes
//   C/D (16x16 f32, 8 VGPR): VGPR r, lane l -> M = r + 8*(l>>4), N = l&15
// Epilogue: +b1, shifted softplus, *W2[n], shfl-reduce over 16 lanes,
// LDS-atomic reduce over waves, +b2, store out[16].
// ---------------------------------------------------------------------------
__global__ __launch_bounds__(512)
void mlp_wmma_kernel(const float* __restrict__ u,
                     const float* __restrict__ W1,   // [128, 256] row-major
                     const float* __restrict__ b1,   // [256]
                     const float* __restrict__ W2,   // [256]
                     const float* __restrict__ b2,   // [1]
                     float*       __restrict__ out)  // [n_graphs]
{
    const int m0   = blockIdx.x * 16;
    const int wave = threadIdx.x >> 5;      // 0..15 -> column tile
    const int lane = threadIdx.x & 31;
    const int half = lane >> 4;             // 0 or 1
    const int l15  = lane & 15;
    const int n0   = wave * 16;

    __shared__ float sums[16];
    if (threadIdx.x < 16) sums[threadIdx.x] = 0.0f;
    __syncthreads();

    // A row this lane feeds (row m0+l15); K pair selected by lane half.
    const float* arow = u + (size_t)(m0 + l15) * HIDDEN;

    v8f acc = {};
    for (int k = 0; k < HIDDEN; k += 4) {
        const int ka = k + half * 2;
        // A fragment: (K=ka, K=ka+1) for row m0+l15 — 8B aligned float2 load.
        v2f a;
        {
            const float2 av = *(const float2*)(arow + ka);
            a.x = av.x; a.y = av.y;
        }
        // B fragment: rows K=ka, K=ka+1 of W1, column n0+l15.
        v2f b;
        b.x = W1[(size_t)ka       * HIDDEN2 + n0 + l15];
        b.y = W1[(size_t)(ka + 1) * HIDDEN2 + n0 + l15];

        acc = __builtin_amdgcn_wmma_f32_16x16x4_f32(
            /*neg_a=*/false, a, /*neg_b=*/false, b,
            /*c_mod=*/(short)0, acc, /*reuse_a=*/false, /*reuse_b=*/false);
    }

    // Epilogue: bias, shifted softplus, W2 dot partials.
    const int   n   = n0 + l15;
    const float bn  = b1[n];
    const float w2n = W2[n];

    float pv[8];
#pragma unroll
    for (int r = 0; r < 8; ++r) {
        const float t  = acc[r] + bn;
        // numerically stable softplus: max(t,0) + log1p(exp(-|t|))
        const float sp = fmaxf(t, 0.0f) + log1pf(expf(-fabsf(t)));
        pv[r] = (sp - SSP_SHIFT) * w2n;
    }

    // Reduce over the 16 lanes of this half (each lane = one N column).
#pragma unroll
    for (int mask = 1; mask < 16; mask <<= 1) {
#pragma unroll
        for (int r = 0; r < 8; ++r)
            pv[r] += __shfl_xor(pv[r], mask, 32);
    }

    // Lane 0 of each half owns 8 row-partials: rows r + 8*half.
    if (l15 == 0) {
#pragma unroll
        for (int r = 0; r < 8; ++r)
            atomicAdd(&sums[r + 8 * half], pv[r]);
    }
    __syncthreads();

    if (threadIdx.x < 16)
        out[m0 + threadIdx.x] = sums[threadIdx.x] + b2[0];
}

// ---------------------------------------------------------------------------
// Launcher
// ---------------------------------------------------------------------------
extern "C" void kernel_launch(void* const* d_in, const int* in_sizes, int n_in,
                              void* d_out, int out_size, void* d_ws, size_t ws_size,
                              hipStream_t stream) {
    const float* v     = (const float*)d_in[0];
    const int*   batch = (const int*)  d_in[1];
    const float* W1    = (const float*)d_in[2];
    const float* b1    = (const float*)d_in[3];
    const float* W2    = (const float*)d_in[4];
    const float* b2    = (const float*)d_in[5];
    float*       out   = (float*)d_out;

    const int n_nodes  = in_sizes[1];     // batch element count
    const int n_graphs = out_size;        // 16384

    float* u = (float*)d_ws;              // [n_graphs, 128] accumulator
    const int u_elems = n_graphs * HIDDEN;

    // 1) zero u
    zero_f32_kernel<<<(u_elems + 255) / 256, 256, 0, stream>>>(u, u_elems);

    // 2) segment sum
    const int nblk = (n_nodes + NODES_PER_BLOCK - 1) / NODES_PER_BLOCK;
    segsum_kernel<<<nblk, 128, 0, stream>>>(v, batch, u, n_nodes);

    // 3) fused WMMA MLP (16 rows per block)
    mlp_wmma_kernel<<<n_graphs / 16, 512, 0, stream>>>(u, W1, b1, W2, b2, out);
}